// SimpleGraphNN_10642928959899
// MI455X (gfx1250) — compile-verified
//
#include <hip/hip_runtime.h>
#include <cstdint>
#include <cstddef>

typedef __attribute__((ext_vector_type(2))) float v2f;
typedef __attribute__((ext_vector_type(8))) float v8f;

// ---------------------------------------------------------------------------
// Layer 1: scatter  agg1[dst] += x[src]*w   (5 floats per edge)
// ---------------------------------------------------------------------------
__global__ void scatter1Kernel(const float* __restrict__ x,
                               const float* __restrict__ eattr,
                               const int* __restrict__ edges,
                               float* __restrict__ agg1, int E) {
    int e = blockIdx.x * blockDim.x + threadIdx.x;
    if (e >= E) return;
    int s = edges[e], d = edges[E + e];
    float w = eattr[2 * e];
#pragma unroll
    for (int k = 0; k < 5; ++k)
        atomicAdd(&agg1[(size_t)d * 5 + k], x[(size_t)s * 5 + k] * w);
}

// ---------------------------------------------------------------------------
// Layer 1 node update: x1 = tanh(agg1@W1_rel + b1 + x@W1_root)   (N x 32)
// ---------------------------------------------------------------------------
__global__ void node1Kernel(const float* __restrict__ x,
                            const float* __restrict__ agg1,
                            const float* __restrict__ W1_rel,
                            const float* __restrict__ b1,
                            const float* __restrict__ W1_root,
                            float* __restrict__ x1, int N) {
    int t = blockIdx.x * blockDim.x + threadIdx.x;
    if (t >= N * 32) return;
    int n = t >> 5, j = t & 31;
    float acc = b1[j];
#pragma unroll
    for (int k = 0; k < 5; ++k)
        acc += agg1[(size_t)n * 5 + k] * W1_rel[k * 32 + j]
             + x[(size_t)n * 5 + k] * W1_root[k * 32 + j];
    x1[t] = tanhf(acc);
}

// ---------------------------------------------------------------------------
// Layer 2: scatter  agg2[dst] += x1[src]*w   (32 floats per edge)
// ---------------------------------------------------------------------------
__global__ void scatter2Kernel(const float* __restrict__ x1,
                               const float* __restrict__ eattr,
                               const int* __restrict__ edges,
                               float* __restrict__ agg2, int E) {
    int e = blockIdx.x * blockDim.x + threadIdx.x;
    if (e >= E) return;
    int s = edges[e], d = edges[E + e];
    float w = eattr[2 * e];
    const float* row = x1 + (size_t)s * 32;
    float* outp = agg2 + (size_t)d * 32;
#pragma unroll
    for (int k = 0; k < 32; ++k) atomicAdd(&outp[k], row[k] * w);
}

// ---------------------------------------------------------------------------
// Layer 2 node update with WMMA:
//   x2 = tanh([agg2 | x1] (N x 64)  @  [W2_rel ; W2_root] (64 x 64)  + b2)
// One wave per 16-node tile; f32 16x16x4 WMMA, K=64 -> 16 ksteps x 4 ntiles.
// ---------------------------------------------------------------------------
__global__ void node2Kernel(const float* __restrict__ agg2,
                            const float* __restrict__ x1,
                            const float* __restrict__ W2_rel,
                            const float* __restrict__ W2_root,
                            const float* __restrict__ b2,
                            float* __restrict__ x2, int nTiles, int N) {
    __shared__ float ldsW[64 * 64];
    int tid = threadIdx.x;
    for (int t = tid; t < 64 * 64; t += 256) {
        int r = t >> 6, c = t & 63;
        ldsW[t] = (r < 32) ? W2_rel[r * 64 + c] : W2_root[(r - 32) * 64 + c];
    }
    __syncthreads();
    int wave = tid >> 5, lane = tid & 31;
    int tile = blockIdx.x * 8 + wave;
    if (tile >= nTiles) return;
    int half = lane >> 4, mrow = lane & 15;
    int node = tile * 16 + mrow;
    if (node >= N) node = N - 1;

    v8f acc[4] = {};
    for (int k = 0; k < 16; ++k) {
        int c = k * 4 + half * 2;
        v2f a;
        a.x = (c < 32) ? agg2[(size_t)node * 32 + c]
                       : x1[(size_t)node * 32 + (c - 32)];
        int c1 = c + 1;
        a.y = (c1 < 32) ? agg2[(size_t)node * 32 + c1]
                        : x1[(size_t)node * 32 + (c1 - 32)];
#pragma unroll
        for (int nt = 0; nt < 4; ++nt) {
            v2f b;
            b.x = ldsW[c * 64 + nt * 16 + mrow];
            b.y = ldsW[c1 * 64 + nt * 16 + mrow];
            acc[nt] = __builtin_amdgcn_wmma_f32_16x16x4_f32(
                false, a, false, b, (short)0, acc[nt], false, false);
        }
    }
    int node0 = tile * 16;
#pragma unroll
    for (int nt = 0; nt < 4; ++nt) {
#pragma unroll
        for (int v = 0; v < 8; ++v) {
            int m = v + half * 8;
            int ncol = nt * 16 + mrow;
            int nn = node0 + m;
            if (nn < N)
                x2[(size_t)nn * 64 + ncol] = tanhf(acc[nt][v] + b2[ncol]);
        }
    }
}

// ---------------------------------------------------------------------------
// Counting sort of kept edges by (e0, e1, origIdx)  ==  stable lexsort
// ---------------------------------------------------------------------------
__global__ void countKernel(const int* __restrict__ edges,
                            const unsigned char* __restrict__ labels,
                            unsigned* __restrict__ cnt, int E) {
    int e = blockIdx.x * blockDim.x + threadIdx.x;
    if (e >= E) return;
    int a = edges[e], b = edges[E + e];
    if (labels[a] && labels[b] && a > b) atomicAdd(&cnt[a], 1u);
}

__global__ void scanKernel(const unsigned* __restrict__ cnt,
                           unsigned* __restrict__ bstart, int n) {
    __shared__ unsigned tmp[1024];
    int tid = threadIdx.x;
    unsigned carry = 0;
    for (int base = 0; base < n; base += 1024) {
        int i = base + tid;
        unsigned v = (i < n) ? cnt[i] : 0u;
        tmp[tid] = v;
        __syncthreads();
        for (int o = 1; o < 1024; o <<= 1) {
            unsigned t = (tid >= o) ? tmp[tid - o] : 0u;
            __syncthreads();
            tmp[tid] += t;
            __syncthreads();
        }
        unsigned incl = tmp[tid];
        if (i < n) bstart[i] = carry + incl - v;
        unsigned total = tmp[1023];
        __syncthreads();
        carry += total;
    }
    if (tid == 0) bstart[n] = carry;
}

__global__ void scatterIdxKernel(const int* __restrict__ edges,
                                 const unsigned char* __restrict__ labels,
                                 const unsigned* __restrict__ bstart,
                                 unsigned* __restrict__ cursor,
                                 unsigned* __restrict__ sidx, int E) {
    int e = blockIdx.x * blockDim.x + threadIdx.x;
    if (e >= E) return;
    int a = edges[e], b = edges[E + e];
    if (labels[a] && labels[b] && a > b) {
        unsigned p = bstart[a] + atomicAdd(&cursor[a], 1u);
        sidx[p] = (unsigned)e;
    }
}

__global__ void bucketSortKernel(const int* __restrict__ edges,
                                 const unsigned* __restrict__ bstart,
                                 const unsigned* __restrict__ cnt,
                                 unsigned* __restrict__ sidx, int N, int E) {
    int n = blockIdx.x * blockDim.x + threadIdx.x;
    if (n >= N) return;
    unsigned s = bstart[n];
    unsigned k = cnt[n];
    for (unsigned i = 1; i < k; ++i) {
        unsigned idx = sidx[s + i];
        unsigned long long key =
            (((unsigned long long)(unsigned)edges[E + idx]) << 32) | idx;
        int j = (int)i - 1;
        while (j >= 0) {
            unsigned idx2 = sidx[s + j];
            unsigned long long key2 =
                (((unsigned long long)(unsigned)edges[E + idx2]) << 32) | idx2;
            if (key2 > key) { sidx[s + j + 1] = idx2; --j; }
            else break;
        }
        sidx[s + (unsigned)(j + 1)] = idx;
    }
}

// ---------------------------------------------------------------------------
// Edge MLP with WMMA:
//   feat = [x2[e0](64) | w | onehot(cls,2) | x2[e1](64)]  (131, padded to 132)
//   h = tanh(feat @ Wd + bd);  efeat = h @ Wo + bo
// One wave per 16-edge tile; Wd (zero-padded) + A tiles staged in LDS.
// ---------------------------------------------------------------------------
__global__ void edgeMlpKernel(const float* __restrict__ x2,
                              const float* __restrict__ Wd,
                              const float* __restrict__ bd,
                              const float* __restrict__ Wo,
                              const float* __restrict__ bo,
                              const int* __restrict__ edges,
                              const float* __restrict__ eattr,
                              const unsigned* __restrict__ sidx,
                              float* __restrict__ efeat,
                              int E, int nTiles, int kept) {
    __shared__ float ldsW[132 * 64];
    __shared__ float ldsA[2][16 * 132];
    int tid = threadIdx.x;
    for (int t = tid; t < 132 * 64; t += 64) {
        int r = t >> 6, c = t & 63;
        ldsW[t] = (r < 131) ? Wd[r * 64 + c] : 0.0f;
    }
    int wave = tid >> 5, lane = tid & 31;
    int half = lane >> 4, mrow = lane & 15;
    int tile = blockIdx.x * 2 + wave;
    if (tile < nTiles) {
        int i = tile * 16 + mrow;
        if (i >= kept) i = kept - 1;
        unsigned j = sidx[i];
        int e0 = edges[j], e1 = edges[E + j];
        float w = eattr[2 * j], cls = eattr[2 * j + 1];
        float* Arow = &ldsA[wave][mrow * 132];
        if (half == 0) {
            const float* r0 = x2 + (size_t)e0 * 64;
            for (int c = 0; c < 64; ++c) Arow[c] = r0[c];
            Arow[64] = w;
            Arow[65] = (cls < 0.5f) ? 1.0f : 0.0f;
        } else {
            const float* r1 = x2 + (size_t)e1 * 64;
            Arow[66] = (cls < 0.5f) ? 0.0f : 1.0f;
            for (int c = 0; c < 64; ++c) Arow[67 + c] = r1[c];
            Arow[131] = 0.0f;
        }
    }
    __syncthreads();
    if (tile >= nTiles) return;

    const float* A = &ldsA[wave][0];
    v8f acc[4] = {};
    for (int k = 0; k < 33; ++k) {
        int c = k * 4 + half * 2;
        v2f a;
        a.x = A[mrow * 132 + c];
        a.y = A[mrow * 132 + c + 1];
#pragma unroll
        for (int nt = 0; nt < 4; ++nt) {
            v2f b;
            b.x = ldsW[c * 64 + nt * 16 + mrow];
            b.y = ldsW[(c + 1) * 64 + nt * 16 + mrow];
            acc[nt] = __builtin_amdgcn_wmma_f32_16x16x4_f32(
                false, a, false, b, (short)0, acc[nt], false, false);
        }
    }
    // h = tanh(D + bd); row-sum of h*Wo across the 64 columns; + bo
#pragma unroll
    for (int v = 0; v < 8; ++v) {
        float rsum = 0.0f;
#pragma unroll
        for (int nt = 0; nt < 4; ++nt) {
            int ncol = nt * 16 + mrow;
            float h = tanhf(acc[nt][v] + bd[ncol]);
            rsum += h * Wo[ncol];
        }
        rsum += __shfl_xor(rsum, 1, 32);
        rsum += __shfl_xor(rsum, 2, 32);
        rsum += __shfl_xor(rsum, 4, 32);
        rsum += __shfl_xor(rsum, 8, 32);
        if (mrow == 0) {
            int m = v + half * 8;
            int i = tile * 16 + m;
            if (i < kept) efeat[i] = rsum + bo[0];
        }
    }
}

// ---------------------------------------------------------------------------
// Pairing: min over (feat[2p], feat[2p+1]) with first-on-tie; emit outputs.
// d_out layout: [e0_even(P) | e1_even(P) | edge_feat(P) | edge_classes(P)]
// ---------------------------------------------------------------------------
__global__ void pairKernel(const int* __restrict__ edges,
                           const float* __restrict__ eattr,
                           const unsigned* __restrict__ sidx,
                           const float* __restrict__ efeat,
                           float* __restrict__ outp, int E, int PAIRS) {
    int p = blockIdx.x * blockDim.x + threadIdx.x;
    if (p >= PAIRS) return;
    unsigned j0 = sidx[2 * p];
    unsigned j1 = sidx[2 * p + 1];
    float f0 = efeat[2 * p], f1 = efeat[2 * p + 1];
    float c0 = eattr[2 * j0 + 1], c1 = eattr[2 * j1 + 1];
    int sel = (f1 < f0) ? 1 : 0;
    outp[p]             = (float)edges[j0];
    outp[PAIRS + p]     = (float)edges[E + j0];
    outp[2 * PAIRS + p] = sel ? f1 : f0;
    outp[3 * PAIRS + p] = sel ? c1 : c0;
}

// ---------------------------------------------------------------------------
extern "C" void kernel_launch(void* const* d_in, const int* in_sizes, int n_in,
                              void* d_out, int out_size, void* d_ws, size_t ws_size,
                              hipStream_t stream) {
    (void)n_in; (void)out_size; (void)ws_size;
    const float* x       = (const float*)d_in[0];
    const float* eattr   = (const float*)d_in[1];
    const float* W1_rel  = (const float*)d_in[2];
    const float* b1      = (const float*)d_in[3];
    const float* W1_root = (const float*)d_in[4];
    const float* W2_rel  = (const float*)d_in[5];
    const float* b2      = (const float*)d_in[6];
    const float* W2_root = (const float*)d_in[7];
    const float* Wd      = (const float*)d_in[8];
    const float* bd      = (const float*)d_in[9];
    const float* Wo      = (const float*)d_in[10];
    const float* bo      = (const float*)d_in[11];
    const int*   edges   = (const int*)d_in[12];
    const unsigned char* labels = (const unsigned char*)d_in[13];

    const int N = in_sizes[0] / 5;
    const int E = in_sizes[12] / 2;
    const int KEPT = E / 2;
    const int PAIRS = KEPT / 2;

    // workspace layout
    char* base = (char*)d_ws;
    size_t off = 0;
    auto take = [&](size_t bytes) -> char* {
        char* p = base + off;
        off += bytes;
        off = (off + 255) & ~(size_t)255;
        return p;
    };
    float*    agg1   = (float*)take((size_t)N * 5 * 4);
    float*    x1     = (float*)take((size_t)N * 32 * 4);
    float*    agg2   = (float*)take((size_t)N * 32 * 4);
    float*    x2     = (float*)take((size_t)N * 64 * 4);
    unsigned* cnt    = (unsigned*)take((size_t)N * 4);
    unsigned* bstart = (unsigned*)take((size_t)(N + 1) * 4);
    unsigned* cursor = (unsigned*)take((size_t)N * 4);
    unsigned* sidx   = (unsigned*)take((size_t)KEPT * 4);
    float*    efeat  = (float*)take((size_t)KEPT * 4);

    hipMemsetAsync(agg1,   0, (size_t)N * 5 * 4,  stream);
    hipMemsetAsync(agg2,   0, (size_t)N * 32 * 4, stream);
    hipMemsetAsync(cnt,    0, (size_t)N * 4,      stream);
    hipMemsetAsync(cursor, 0, (size_t)N * 4,      stream);

    const int TB = 256;
    // layer 1
    scatter1Kernel<<<(E + TB - 1) / TB, TB, 0, stream>>>(x, eattr, edges, agg1, E);
    node1Kernel<<<((size_t)N * 32 + TB - 1) / TB, TB, 0, stream>>>(
        x, agg1, W1_rel, b1, W1_root, x1, N);
    // layer 2
    scatter2Kernel<<<(E + TB - 1) / TB, TB, 0, stream>>>(x1, eattr, edges, agg2, E);
    int nTiles = (N + 15) / 16;
    node2Kernel<<<(nTiles + 7) / 8, 256, 0, stream>>>(
        agg2, x1, W2_rel, W2_root, b2, x2, nTiles, N);
    // split + lexsort (counting sort on e0, per-bucket sort on (e1, idx))
    countKernel<<<(E + TB - 1) / TB, TB, 0, stream>>>(edges, labels, cnt, E);
    scanKernel<<<1, 1024, 0, stream>>>(cnt, bstart, N);
    scatterIdxKernel<<<(E + TB - 1) / TB, TB, 0, stream>>>(
        edges, labels, bstart, cursor, sidx, E);
    bucketSortKernel<<<(N + TB - 1) / TB, TB, 0, stream>>>(
        edges, bstart, cnt, sidx, N, E);
    // edge MLP (WMMA)
    int nTiles2 = (KEPT + 15) / 16;
    edgeMlpKernel<<<(nTiles2 + 1) / 2, 64, 0, stream>>>(
        x2, Wd, bd, Wo, bo, edges, eattr, sidx, efeat, E, nTiles2, KEPT);
    // pairing + outputs
    pairKernel<<<(PAIRS + TB - 1) / TB, TB, 0, stream>>>(
        edges, eattr, sidx, efeat, (float*)d_out, E, PAIRS);
}